// MaxRecallLoss_23218593202556
// MI455X (gfx1250) — compile-verified
//
#include <hip/hip_runtime.h>
#include <hip/hip_bf16.h>
#include <math.h>

typedef __attribute__((ext_vector_type(2))) float v2f;
typedef __attribute__((ext_vector_type(8))) float v8f;

#define NB  512   // partial-sum blocks
#define TPB 256   // 8 waves per block (wave32)

#define LOG2E  1.4426950408889634f
#define LN2    0.6931471805599453f
#define INVT   (1.0f / 1.5f)

// Wave32 all-lane sum on the matrix pipe (f32 end-to-end).
// A(16x4): both A VGPRs = v -> row m of A holds {v[m], v[m], v[m+16], v[m+16]}
// B(4x16): all ones         -> D[m][n] = 2*(v[m] + v[m+16]) for every n
// Summing the 8 D VGPRs gives 2*Sigma(lo-half rows) in lanes 0-15 and
// 2*Sigma(hi-half rows) in lanes 16-31; one xor-16 shuffle + 0.5x finishes.
__device__ inline float wave_reduce_wmma(float v) {
  v2f a; a[0] = v;    a[1] = v;
  v2f b; b[0] = 1.0f; b[1] = 1.0f;
  v8f c = {};
  v8f d = __builtin_amdgcn_wmma_f32_16x16x4_f32(
      false, a, false, b, (short)0, c, false, false);
  float s = ((d[0] + d[1]) + (d[2] + d[3])) + ((d[4] + d[5]) + (d[6] + d[7]));
  s = 0.5f * (s + __shfl_xor(s, 16, 32));
  return s;
}

__global__ void __launch_bounds__(TPB)
MaxRecallLoss_partials(const float* __restrict__ logits,
                       const int*   __restrict__ targets,
                       const float* __restrict__ counts,
                       float* __restrict__ ws, int B) {
  // base class weights (1/sqrt(cc+1), normalized to sum to C=8) -> LDS table
  __shared__ float sbw[8];
  if (threadIdx.x == 0) {
    float w[8]; float wsum = 0.f;
#pragma unroll
    for (int c = 0; c < 8; ++c) { w[c] = rsqrtf(counts[c] + 1.0f); wsum += w[c]; }
    const float scale = 8.0f / wsum;
#pragma unroll
    for (int c = 0; c < 8; ++c) sbw[c] = w[c] * scale;
  }
  __syncthreads();

  float ce_acc = 0.f, tp_acc = 0.f;
  unsigned cnt_bits = 0u;                       // cancer-target count (SALU pipe)
  const int stride = gridDim.x * blockDim.x;
  for (int row = blockIdx.x * blockDim.x + threadIdx.x; row < B; row += stride) {
    const float4* p = (const float4*)(logits + (size_t)row * 8);
    __builtin_prefetch(logits + ((size_t)row + (size_t)stride) * 8, 0, 0);
    float4 lo = p[0], hi = p[1];
    float x[8] = {lo.x, lo.y, lo.z, lo.w, hi.x, hi.y, hi.z, hi.w};
    const int t = targets[row];
    const float bwt = sbw[t];

    // argmax, first-max tie-break (matches jnp.argmax)
    int pred = 0; float best = x[0];
#pragma unroll
    for (int c = 1; c < 8; ++c) if (x[c] > best) { best = x[c]; pred = c; }

    // shifted logits (shared by both softmaxes)
    float u[8];
#pragma unroll
    for (int c = 0; c < 8; ++c) u[c] = x[c] - best;

    float seT = 0.f, se1 = 0.f, e1[8];
#pragma unroll
    for (int c = 0; c < 8; ++c) {
      seT += __builtin_amdgcn_exp2f(u[c] * (LOG2E * INVT));  // temp-1.5 softmax
      e1[c] = __builtin_amdgcn_exp2f(u[c] * LOG2E);          // temp-1 softmax
      se1 += e1[c];
    }
    const float log2seT = __builtin_amdgcn_logf(seT);

    float sx = 0.f;
#pragma unroll
    for (int c = 0; c < 8; ++c) sx += x[c];
    const float cx = x[0] + x[1] + x[3];
    float xt = 0.f;
#pragma unroll
    for (int c = 0; c < 8; ++c) xt = (t == c) ? x[c] : xt;

    const bool isct = (t == 0) || (t == 1) || (t == 3);
    // smoothed-label dot: cancer rows sum to 1; benign rows pre-norm sum = 1.05
    const float ka = isct ? 0.00625f : (0.0125f / 1.05f);        // smooth/C
    const float kt = isct ? 0.95f    : (0.9f / 1.05f);           // 1-smooth
    const float kc = isct ? 0.0f     : (0.1f / 6.0f / 1.05f);    // extra on cancer cls
    const float sdotx = ka * sx + kt * xt + kc * cx;
    float ce = INVT * (best - sdotx) + LN2 * log2seT;            // logZ - s.y

    const bool iscp = (pred == 0) || (pred == 1) || (pred == 3);
    float mult = 1.0f;
    if (isct && !iscp)       mult = 3.0f;   // FN_MULT
    if ((t == 0) && !iscp)   mult = 5.0f;   // MEL_FN_MULT (overrides)
    if (isct && (pred != t)) mult *= 2.0f;  // HARD_WEIGHT
    ce_acc += ce * bwt * mult;

    // cancer probability mass at temp 1 (fast rcp instead of IEEE divide)
    const float cp = (e1[0] + e1[1] + e1[3]) * __builtin_amdgcn_rcpf(se1);
    tp_acc += isct ? cp : 0.0f;
    cnt_bits += __popc((unsigned)__ballot(isct));
  }

  // wave32 totals on the matrix pipe, then cross-wave via LDS
  const float ce_w = wave_reduce_wmma(ce_acc);
  const float tp_w = wave_reduce_wmma(tp_acc);

  __shared__ float sce[TPB / 32], stp[TPB / 32], scnt[TPB / 32];
  const int wid = threadIdx.x >> 5, lid = threadIdx.x & 31;
  if (lid == 0) { sce[wid] = ce_w; stp[wid] = tp_w; scnt[wid] = (float)cnt_bits; }
  __syncthreads();
  if (threadIdx.x == 0) {
    float a = 0.f, b = 0.f, c = 0.f;
#pragma unroll
    for (int i = 0; i < TPB / 32; ++i) { a += sce[i]; b += stp[i]; c += scnt[i]; }
    ws[blockIdx.x]          = a;
    ws[NB + blockIdx.x]     = b;
    ws[2 * NB + blockIdx.x] = c;
  }
}

__global__ void __launch_bounds__(TPB)
MaxRecallLoss_finalize(const float* __restrict__ ws, float* __restrict__ out, int B) {
  float ce = 0.f, tp = 0.f, cnt = 0.f;
  for (int i = threadIdx.x; i < NB; i += TPB) {
    ce += ws[i]; tp += ws[NB + i]; cnt += ws[2 * NB + i];
  }
  ce  = wave_reduce_wmma(ce);
  tp  = wave_reduce_wmma(tp);
  cnt = wave_reduce_wmma(cnt);

  __shared__ float sce[TPB / 32], stp[TPB / 32], scnt[TPB / 32];
  const int wid = threadIdx.x >> 5, lid = threadIdx.x & 31;
  if (lid == 0) { sce[wid] = ce; stp[wid] = tp; scnt[wid] = cnt; }
  __syncthreads();
  if (threadIdx.x == 0) {
    float a = 0.f, b = 0.f, c = 0.f;
#pragma unroll
    for (int i = 0; i < TPB / 32; ++i) { a += sce[i]; b += stp[i]; c += scnt[i]; }
    const float base_loss   = a / (float)B;
    const float soft_recall = b / (c + 1e-8f);   // tp + fn == cancer count
    out[0] = base_loss + 0.5f * (1.0f - soft_recall);
  }
}

extern "C" void kernel_launch(void* const* d_in, const int* in_sizes, int n_in,
                              void* d_out, int out_size, void* d_ws, size_t ws_size,
                              hipStream_t stream) {
  const float* logits  = (const float*)d_in[0];
  const int*   targets = (const int*)d_in[1];
  const float* counts  = (const float*)d_in[2];
  const int B = in_sizes[1];           // number of rows (targets is [B])
  float* ws = (float*)d_ws;            // 3*NB floats of block partials

  MaxRecallLoss_partials<<<NB, TPB, 0, stream>>>(logits, targets, counts, ws, B);
  MaxRecallLoss_finalize<<<1, TPB, 0, stream>>>(ws, (float*)d_out, B);
}